// MNISTNet_67602785239194
// MI455X (gfx1250) — compile-verified
//
#include <hip/hip_runtime.h>
#include <math.h>

// Problem constants (fixed shapes from the reference)
#define GGR   1024          // graphs
#define PN    75            // nodes per graph
#define PPAD  80            // padded to multiple of 16 for WMMA M tiles
#define EPG   600           // edges per graph (75 * 8)
#define PAIRS 2400          // (edge, spline-corner) pairs per graph
#define KKER  25            // spline kernels
#define NN    (GGR * PN)    // 76800 nodes
#define EE    (NN * 8)      // 614400 edges

typedef __attribute__((ext_vector_type(2))) float v2f;
typedef __attribute__((ext_vector_type(8))) float v8f;

__device__ __forceinline__ float elu1(float x) { return x > 0.f ? x : (expf(x) - 1.f); }

// ---------------------------------------------------------------------------
// Kernel 0: B-spline basis/index per edge + in-degree per node
// ---------------------------------------------------------------------------
__global__ __launch_bounds__(256)
void k_basis(const float* __restrict__ ea, const int* __restrict__ dst,
             float* __restrict__ basis, int* __restrict__ idx, int* __restrict__ deg)
{
    int e = blockIdx.x * blockDim.x + threadIdx.x;
    if (e >= EE) return;
    float p0 = ea[e * 2 + 0] * 4.f, p1 = ea[e * 2 + 1] * 4.f;
    float l0 = fminf(fmaxf(floorf(p0), 0.f), 4.f);
    float l1 = fminf(fmaxf(floorf(p1), 0.f), 4.f);
    float f0 = p0 - l0, f1 = p1 - l1;
    int i0 = (int)l0, i1 = (int)l1;
    int h0 = min(i0 + 1, 4), h1 = min(i1 + 1, 4);
    // s = s0*2 + s1 ; basis = w0[s0]*w1[s1] ; idx = i0[s0] + 5*i1[s1]
    basis[e * 4 + 0] = (1.f - f0) * (1.f - f1); idx[e * 4 + 0] = i0 + 5 * i1;
    basis[e * 4 + 1] = (1.f - f0) * f1;         idx[e * 4 + 1] = i0 + 5 * h1;
    basis[e * 4 + 2] = f0 * (1.f - f1);         idx[e * 4 + 2] = h0 + 5 * i1;
    basis[e * 4 + 3] = f0 * f1;                 idx[e * 4 + 3] = h0 + 5 * h1;
    atomicAdd(&deg[dst[e]], 1);
}

// ---------------------------------------------------------------------------
// Fused SplineConv layer: one graph per workgroup, everything in LDS.
//   agg[d,:] = sum_k xk[d,k,:] @ W[k]       (WMMA f32 16x16x4, K chunked)
//   out      = elu(agg/deg + h@root + bias) (root term also via WMMA)
// 320 threads = 10 wave32; 5 M-tiles x (COUT/16) N-tiles, <=2 tiles per wave.
// ---------------------------------------------------------------------------
template<int CIN, int COUT, int KCHUNK, int NCHUNKS, int HSTRIDE, int XSTRIDE>
__global__ __launch_bounds__(320)
void k_spline(const float* __restrict__ h_in, const float* __restrict__ W,
              const float* __restrict__ root, const float* __restrict__ bias,
              const float* __restrict__ basis, const int* __restrict__ idx,
              const int* __restrict__ dst, const int* __restrict__ deg,
              float* __restrict__ h_out)
{
    constexpr int KP     = (KCHUNK * CIN + 3) & ~3;  // padded GEMM K per chunk
    constexpr int KR     = (CIN + 3) & ~3;           // padded GEMM K for root
    constexpr int NTILES = 5 * (COUT / 16);
    static_assert(XSTRIDE >= KP && HSTRIDE >= KR && NTILES <= 20, "tiling");

    __shared__ float         sh_h [PPAD * HSTRIDE];  // staged input features
    __shared__ float         sh_xk[PPAD * XSTRIDE];  // per-chunk scattered xk
    __shared__ float         sh_b [PAIRS];           // basis per (edge,corner)
    __shared__ unsigned char sh_k [PAIRS];           // spline index per pair
    __shared__ unsigned char sh_d [EPG];             // local dst per edge

    const int g    = blockIdx.x;
    const int tid  = threadIdx.x;
    const int wid  = tid >> 5;
    const int lane = tid & 31;

    // ---- stage graph-local data into LDS ----
    for (int j = tid; j < PPAD * HSTRIDE; j += 320) sh_h[j] = 0.f;
    __syncthreads();
    for (int j = tid; j < PN * CIN; j += 320)
        sh_h[(j / CIN) * HSTRIDE + (j % CIN)] = h_in[g * PN * CIN + j];
    for (int p = tid; p < PAIRS; p += 320) {
        sh_b[p] = basis[g * PAIRS + p];
        sh_k[p] = (unsigned char)idx[g * PAIRS + p];
    }
    for (int q = tid; q < EPG; q += 320)
        sh_d[q] = (unsigned char)(dst[g * EPG + q] - g * PN);
    __syncthreads();

    v8f acc0 = {}; v8f acc1 = {};   // message accumulators live across chunks

    for (int c = 0; c < NCHUNKS; ++c) {
        const int k0 = c * KCHUNK;
        for (int j = tid; j < PPAD * XSTRIDE; j += 320) sh_xk[j] = 0.f;
        __syncthreads();

        // scatter: one (edge, corner) pair per wave, lanes cover CIN channels
        for (int p = wid; p < PAIRS; p += 10) {
            int k = sh_k[p];
            if (k >= k0 && k < k0 + KCHUNK) {
                float bsc = sh_b[p];
                int srcl = p >> 5;           // src = edge/8, edge = p/4
                int dstl = sh_d[p >> 2];
                float*       dp = &sh_xk[dstl * XSTRIDE + (k - k0) * CIN];
                const float* sp = &sh_h [srcl * HSTRIDE];
                for (int i = lane; i < CIN; i += 32)
                    atomicAdd(&dp[i], bsc * sp[i]);    // ds_add_f32
            }
        }
        __syncthreads();

        // GEMM-accumulate this chunk: [80 x KCHUNK*CIN] x [KCHUNK*CIN x COUT]
        for (int ti = 0; ti < 2; ++ti) {
            int t = wid + ti * 10;
            if (t >= NTILES) break;                      // wave-uniform
            int mt = t % 5, nt = t / 5;
            v8f accl = ti ? acc1 : acc0;
            const int arow = (mt * 16 + (lane & 15)) * XSTRIDE;
            const int n    =  nt * 16 + (lane & 15);
            const int koff = 2 * (lane >> 4);
            for (int kk = 0; kk < KP; kk += 4) {
                v2f a, b;
                a.x = sh_xk[arow + kk + koff];
                a.y = sh_xk[arow + kk + koff + 1];
                int r0 = kk + koff;
                b.x = (r0     < KCHUNK * CIN) ? W[(k0 * CIN + r0    ) * COUT + n] : 0.f;
                b.y = (r0 + 1 < KCHUNK * CIN) ? W[(k0 * CIN + r0 + 1) * COUT + n] : 0.f;
                accl = __builtin_amdgcn_wmma_f32_16x16x4_f32(
                    false, a, false, b, (short)0, accl, false, false);
            }
            if (ti) acc1 = accl; else acc0 = accl;
        }
        __syncthreads();
    }

    // ---- root-weight GEMM + mean-normalize + bias + ELU + store ----
    for (int ti = 0; ti < 2; ++ti) {
        int t = wid + ti * 10;
        if (t >= NTILES) break;                          // wave-uniform
        int mt = t % 5, nt = t / 5;
        v8f msg = ti ? acc1 : acc0;
        v8f rt  = {};
        const int arow = (mt * 16 + (lane & 15)) * HSTRIDE;
        const int n    =  nt * 16 + (lane & 15);
        const int koff = 2 * (lane >> 4);
        for (int kk = 0; kk < KR; kk += 4) {
            v2f a, b;
            a.x = sh_h[arow + kk + koff];
            a.y = sh_h[arow + kk + koff + 1];
            int r0 = kk + koff;
            b.x = (r0     < CIN) ? root[(r0    ) * COUT + n] : 0.f;
            b.y = (r0 + 1 < CIN) ? root[(r0 + 1) * COUT + n] : 0.f;
            rt = __builtin_amdgcn_wmma_f32_16x16x4_f32(
                false, a, false, b, (short)0, rt, false, false);
        }
        float bv = bias[n];
        for (int r = 0; r < 8; ++r) {                    // C layout: M=r+8*(lane/16)
            int ml = mt * 16 + r + 8 * (lane >> 4);
            if (ml < PN) {
                int node = g * PN + ml;
                float dg = (float)max(deg[node], 1);
                float v  = msg[r] / dg + rt[r] + bv;
                h_out[node * COUT + n] = elu1(v);
            }
        }
    }
}

// ---------------------------------------------------------------------------
// Head: voxel max-pool (2x2 per graph) -> fc1(256,128)+elu -> fc2(128,10)
//       -> log_softmax.  One graph per block, 256 threads.
// ---------------------------------------------------------------------------
__global__ __launch_bounds__(256)
void k_head(const float* __restrict__ h3, const float* __restrict__ pos,
            const float* __restrict__ f1w, const float* __restrict__ f1b,
            const float* __restrict__ f2w, const float* __restrict__ f2b,
            float* __restrict__ out)
{
    __shared__ int   sh_cl[PN];
    __shared__ float sh_gx[256];
    __shared__ float sh_a1[128];
    __shared__ float sh_lg[10];
    __shared__ float sh_ls;
    int g = blockIdx.x, tid = threadIdx.x;

    for (int n = tid; n < PN; n += 256) {
        int node = g * PN + n;
        int v0 = min(max((int)(pos[node * 2 + 0] / 14.f), 0), 1);
        int v1 = min(max((int)(pos[node * 2 + 1] / 14.f), 0), 1);
        sh_cl[n] = v0 + 2 * v1;
    }
    __syncthreads();
    {
        int v = tid >> 6, o = tid & 63;
        float m = -INFINITY;
        for (int n = 0; n < PN; ++n)
            if (sh_cl[n] == v) m = fmaxf(m, h3[(g * PN + n) * 64 + o]);
        sh_gx[tid] = (m == -INFINITY) ? 0.f : m;   // empty voxel -> 0
    }
    __syncthreads();
    if (tid < 128) {
        float s = f1b[tid];
        for (int i = 0; i < 256; ++i) s += sh_gx[i] * f1w[i * 128 + tid];
        sh_a1[tid] = elu1(s);
    }
    __syncthreads();
    if (tid < 10) {
        float s = f2b[tid];
        for (int i = 0; i < 128; ++i) s += sh_a1[i] * f2w[i * 10 + tid];
        sh_lg[tid] = s;
    }
    __syncthreads();
    if (tid == 0) {
        float mx = sh_lg[0];
        for (int c = 1; c < 10; ++c) mx = fmaxf(mx, sh_lg[c]);
        float se = 0.f;
        for (int c = 0; c < 10; ++c) se += expf(sh_lg[c] - mx);
        sh_ls = mx + logf(se);
    }
    __syncthreads();
    if (tid < 10) out[g * 10 + tid] = sh_lg[tid] - sh_ls;
}

// ---------------------------------------------------------------------------
extern "C" void kernel_launch(void* const* d_in, const int* in_sizes, int n_in,
                              void* d_out, int out_size, void* d_ws, size_t ws_size,
                              hipStream_t stream)
{
    const float* x   = (const float*)d_in[0];
    const int*   ei  = (const int*)  d_in[1];
    const float* ea  = (const float*)d_in[2];
    const float* pos = (const float*)d_in[4];
    const float* W1  = (const float*)d_in[5];
    const float* r1  = (const float*)d_in[6];
    const float* b1  = (const float*)d_in[7];
    const float* W2  = (const float*)d_in[8];
    const float* r2  = (const float*)d_in[9];
    const float* b2  = (const float*)d_in[10];
    const float* W3  = (const float*)d_in[11];
    const float* r3  = (const float*)d_in[12];
    const float* b3  = (const float*)d_in[13];
    const float* f1w = (const float*)d_in[14];
    const float* f1b = (const float*)d_in[15];
    const float* f2w = (const float*)d_in[16];
    const float* f2b = (const float*)d_in[17];
    const int*   dst = ei + EE;                 // edge_index row 1

    // scratch carve-out (~69 MB)
    size_t off = 0;
    auto carve = [&](size_t bytes) -> void* {
        off = (off + 255) & ~(size_t)255;
        void* p = (char*)d_ws + off;
        off += bytes;
        return p;
    };
    float* basis = (float*)carve((size_t)EE * 4 * sizeof(float));
    int*   idx   = (int*)  carve((size_t)EE * 4 * sizeof(int));
    int*   deg   = (int*)  carve((size_t)NN * sizeof(int));
    float* h1    = (float*)carve((size_t)NN * 32 * sizeof(float));
    float* h2    = (float*)carve((size_t)NN * 64 * sizeof(float));
    float* h3    = (float*)carve((size_t)NN * 64 * sizeof(float));

    hipMemsetAsync(deg, 0, (size_t)NN * sizeof(int), stream);
    k_basis<<<(EE + 255) / 256, 256, 0, stream>>>(ea, dst, basis, idx, deg);

    // <CIN, COUT, KCHUNK, NCHUNKS, HSTRIDE, XSTRIDE>  (strides = 4*odd floats)
    k_spline<1,  32, 25, 1,  4,  28><<<GGR, 320, 0, stream>>>(x,  W1, r1, b1, basis, idx, dst, deg, h1);
    k_spline<32, 64,  5, 5, 36, 164><<<GGR, 320, 0, stream>>>(h1, W2, r2, b2, basis, idx, dst, deg, h2);
    k_spline<64, 64,  5, 5, 68, 324><<<GGR, 320, 0, stream>>>(h2, W3, r3, b3, basis, idx, dst, deg, h3);

    k_head<<<GGR, 256, 0, stream>>>(h3, pos, f1w, f1b, f2w, f2b, (float*)d_out);
}